// HierarchicalLocationEmbedding_52063593562206
// MI455X (gfx1250) — compile-verified
//
#include <hip/hip_runtime.h>

// HierarchicalLocationEmbedding fused kernel for gfx1250 (MI455X).
// B=8, S=2048 -> T=16384 tokens. VOCAB=100000, H/2=256, NUM_CLUSTERS=256.
//
// Memory-bound: ~150MB touched first pass (~6.5us @ 23.3TB/s); loc_to_cluster_w
// (102MB) and coarse_embed_w (256KB) are L2-resident (192MB) on replay.
// GEMM is 2.1 GFLOP -> done exactly in fp32 with V_WMMA_F32_16X16X4_F32.

#define VOCABN 100000
#define HH 256          // HIDDEN/2
#define NC 256          // clusters
#define TOK_PER_BLK 16
#define PAD 17          // LDS row pad: bank-conflict-free for stride-16 scatter
                        // and for A-fragment reads (17*s mod 64 distinct, s=0..15)

typedef float v2f __attribute__((ext_vector_type(2)));
typedef float v8f __attribute__((ext_vector_type(8)));

__global__ __launch_bounds__(256)
void hle_kernel(const int* __restrict__ ids,
                const float* __restrict__ fine_w,     // [VOCAB, 256]
                const float* __restrict__ coarse_w,   // [256, 256]
                const float* __restrict__ loc_w,      // [256, VOCAB]
                const float* __restrict__ loc_b,      // [256]
                float* __restrict__ out)              // [T, 512]
{
    __shared__ float distT[NC * PAD];   // distT[k*PAD + m], m = token-in-block

    const int tid   = threadIdx.x;
    const int token = tid >> 4;         // 0..15  (16 threads per token)
    const int sub   = tid & 15;         // 0..15
    const int g     = blockIdx.x * TOK_PER_BLK + token;
    const int id    = ids[g];

    // ---------- Phase 1: strided column gather of loc_w + bias, softmax ----------
    float l[16];
    #pragma unroll
    for (int j = 0; j < 16; ++j) {
        const int c = j * 16 + sub;
        l[j] = loc_w[(size_t)c * VOCABN + id] + loc_b[c];
    }
    float mx = l[0];
    #pragma unroll
    for (int j = 1; j < 16; ++j) mx = fmaxf(mx, l[j]);
    // reduce across the contiguous 16-lane group (wave32: groups never straddle a wave)
    #pragma unroll
    for (int off = 1; off < 16; off <<= 1) mx = fmaxf(mx, __shfl_xor(mx, off, 32));
    float s = 0.f;
    #pragma unroll
    for (int j = 0; j < 16; ++j) { l[j] = __expf(l[j] - mx); s += l[j]; }
    #pragma unroll
    for (int off = 1; off < 16; off <<= 1) s += __shfl_xor(s, off, 32);
    const float inv = 1.f / s;
    #pragma unroll
    for (int j = 0; j < 16; ++j) {
        const int c = j * 16 + sub;
        distT[c * PAD + token] = l[j] * inv;   // transposed store [k][m]
    }

    // ---------- Phase 1b: fine embedding gather (b128 coalesced) ----------
    {
        const float4* src = (const float4*)(fine_w + (size_t)id * HH + sub * 16);
        float4*       dst = (float4*)(out + (size_t)g * 512 + sub * 16);
        #pragma unroll
        for (int q = 0; q < 4; ++q) dst[q] = src[q];
    }

    __syncthreads();

    // ---------- Phase 2: coarse = dist @ coarse_w via V_WMMA_F32_16X16X4_F32 ----
    // A 16x4 f32 layout (ISA 7.12.2): lanes 0-15 hold K=k0,k0+1; lanes 16-31 K=k0+2,k0+3.
    // B 4x16 assumed mirrored K-split across lane halves (row striped across lanes).
    const int wave  = tid >> 5;          // 0..7, each wave -> two 16-col N tiles
    const int lane  = tid & 31;
    const int mrow  = lane & 15;
    const int khalf = (lane >> 4) * 2;

    #pragma unroll
    for (int rep = 0; rep < 2; ++rep) {
        const int n0  = (wave * 2 + rep) * 16;
        const int col = n0 + mrow;
        v8f acc = {};
        #pragma unroll 4
        for (int k0 = 0; k0 < NC; k0 += 4) {
            const int k = k0 + khalf;
            v2f a, b;
            a.x = distT[k * PAD + mrow];
            a.y = distT[(k + 1) * PAD + mrow];
            b.x = coarse_w[(size_t)k * HH + col];        // L2-resident (256KB)
            b.y = coarse_w[(size_t)(k + 1) * HH + col];
            acc = __builtin_amdgcn_wmma_f32_16x16x4_f32(
                /*neg_a=*/false, a, /*neg_b=*/false, b,
                /*c_mod=*/(short)0, acc, /*reuse_a=*/false, /*reuse_b=*/false);
        }
        // C/D layout: VGPR v -> M = v + 8*(lane>=16); N = lane&15
        const int mbase = (lane >> 4) * 8;
        float* orow = out + (size_t)blockIdx.x * TOK_PER_BLK * 512 + 256 + col;
        #pragma unroll
        for (int v = 0; v < 8; ++v)
            orow[(size_t)(mbase + v) * 512] = acc[v];
    }
}

extern "C" void kernel_launch(void* const* d_in, const int* in_sizes, int n_in,
                              void* d_out, int out_size, void* d_ws, size_t ws_size,
                              hipStream_t stream) {
    const int*   ids      = (const int*)  d_in[0];   // location_ids [B,S]
    const float* fine_w   = (const float*)d_in[1];   // [VOCAB, 256]
    const float* coarse_w = (const float*)d_in[2];   // [256, 256]
    const float* loc_w    = (const float*)d_in[3];   // [256, VOCAB]
    const float* loc_b    = (const float*)d_in[4];   // [256]
    float* out = (float*)d_out;

    const int T = in_sizes[0];                        // 16384 tokens
    hle_kernel<<<T / TOK_PER_BLK, 256, 0, stream>>>(ids, fine_w, coarse_w,
                                                    loc_w, loc_b, out);
}